// LatentLookup_88338887344155
// MI455X (gfx1250) — compile-verified
//
#include <hip/hip_runtime.h>
#include <math.h>

typedef float v2f __attribute__((ext_vector_type(2)));
typedef float v8f __attribute__((ext_vector_type(8)));

#define WAVES_PER_BLOCK 8
#define TILES_PER_WAVE  256
#define UNROLL          4
// BIAS > max |q|^2 (~67.7) so D = |p|^2 - 2 q.p + BIAS > 0 always:
// positive IEEE floats compare correctly as raw unsigned bits.
#define D2_BIAS         128.0f

// ---------------------------------------------------------------------------
// Workspace init: per-query packed (key<<32 | idx) argmin slots -> all ones.
// ---------------------------------------------------------------------------
__global__ void ll_init(unsigned long long* __restrict__ ws, int n) {
  int i = blockIdx.x * blockDim.x + threadIdx.x;
  if (i < n) ws[i] = 0xFFFFFFFFFFFFFFFFull;
}

// ---------------------------------------------------------------------------
// Hard argmin over the full database via V_WMMA_F32_16X16X4_F32.
// Each wave owns one 16-query tile and a chunk of 256 16-point column tiles.
// D[m][n] = |p_n|^2 - 2 q_m . p_n + BIAS   (|q|^2 dropped: argmin-invariant;
// BIAS keeps D positive so raw f32 bits are an unsigned-monotone key).
// A = -2*Q (16x4, K=(x,y,z,0)), C = |p_n|^2 + BIAS broadcast across rows.
// ---------------------------------------------------------------------------
__global__ void __launch_bounds__(256) ll_hard_wmma(
    const float* __restrict__ q,          // [B,3]
    const float* __restrict__ db,         // [ndb,3]  (indices_3d flattened)
    unsigned long long* __restrict__ ws,  // [B] packed argmin slots
    int B, int ndb)
{
  const int lane   = (int)(threadIdx.x & 31u);
  const int wave   = (int)(threadIdx.x >> 5u);
  const int qt     = blockIdx.y * WAVES_PER_BLOCK + wave;   // query tile id
  if (qt * 16 >= B) return;                                 // wave-uniform

  const bool hi     = lane >= 16;
  const int  lane15 = lane & 15;
  const int  qrow   = qt * 16 + lane15;

  float qx = 0.f, qy = 0.f, qz = 0.f;
  if (qrow < B) { qx = q[qrow*3+0]; qy = q[qrow*3+1]; qz = q[qrow*3+2]; }
  v2f a;                            // A pre-scaled by -2
  a.x = -2.0f * (hi ? qz : qx);     // K=2 : K=0
  a.y = -2.0f * (hi ? 0.f : qy);    // K=3 : K=1

  unsigned long long best[8];
#pragma unroll
  for (int r = 0; r < 8; ++r) best[r] = 0xFFFFFFFFFFFFFFFFull;

  const int t0 = blockIdx.x * TILES_PER_WAVE;
  for (int t = 0; t < TILES_PER_WAVE; t += UNROLL) {
    float bx[UNROLL], by[UNROLL], bz[UNROLL];
    int   pc[UNROLL];
    bool  ok[UNROLL];
#pragma unroll
    for (int u = 0; u < UNROLL; ++u) {
      int p = (t0 + t + u) * 16 + lane15;       // db point index for this lane
      ok[u] = p < ndb;
      pc[u] = ok[u] ? p : (ndb - 1);
      bx[u] = db[(size_t)pc[u]*3+0];
      by[u] = db[(size_t)pc[u]*3+1];
      bz[u] = db[(size_t)pc[u]*3+2];
    }
    {
      // stay ahead of the stream (cold-L2 first pass): global_prefetch_b8
      int pf = (t0 + t + 64) * 16 + lane15;
      pf = pf < ndb ? pf : (ndb - 1);
      __builtin_prefetch(&db[(size_t)pf*3], 0, 0);
    }
#pragma unroll
    for (int u = 0; u < UNROLL; ++u) {
      float inorm = bx[u]*bx[u] + by[u]*by[u] + bz[u]*bz[u] + D2_BIAS;
      if (!ok[u]) inorm = 3.0e38f;

      v2f bm;
      bm.x = hi ? bz[u] : bx[u];               // K=2 : K=0
      bm.y = hi ? 0.f   : by[u];               // K=3 : K=1

      v8f c;
#pragma unroll
      for (int r = 0; r < 8; ++r) c[r] = inorm;  // C broadcast across rows

      v8f d = __builtin_amdgcn_wmma_f32_16x16x4_f32(
          false, a, false, bm, (short)0, c, false, false);

      unsigned long long idx = (unsigned long long)(unsigned int)pc[u];
#pragma unroll
      for (int r = 0; r < 8; ++r) {
        // D > 0 => raw bits are unsigned-monotone; ties -> smallest index.
        unsigned long long cand =
            ((unsigned long long)__float_as_uint(d[r]) << 32) | idx;
        best[r] = (cand < best[r]) ? cand : best[r];
      }
    }
  }

  // Butterfly min within each 16-lane half (columns of a fixed row group).
#pragma unroll
  for (int r = 0; r < 8; ++r) {
    unsigned long long v = best[r];
    for (int m = 1; m < 16; m <<= 1) {
      unsigned int lo = (unsigned int)v;
      unsigned int hw = (unsigned int)(v >> 32);
      lo = (unsigned int)__shfl_xor((int)lo, m, 32);
      hw = (unsigned int)__shfl_xor((int)hw, m, 32);
      unsigned long long o = ((unsigned long long)hw << 32) | lo;
      v = (o < v) ? o : v;
    }
    best[r] = v;
  }

  if (lane15 == 0) {
#pragma unroll
    for (int r = 0; r < 8; ++r) {
      int row = qt * 16 + r + (hi ? 8 : 0);
      if (row < B) atomicMin(&ws[row], best[r]);
    }
  }
}

// ---------------------------------------------------------------------------
// Finalize: gather hard metrics + exact 27-neighbor softmax (soft_get_local).
// ---------------------------------------------------------------------------
__global__ void ll_finalize(
    const float* __restrict__ q,        // [B,3]
    const float* __restrict__ temp_p,   // [1]
    const float* __restrict__ grid,     // [n,n,n,3]
    const float* __restrict__ met,      // [n,n,n,2]
    const float* __restrict__ origin,   // [3]
    const float* __restrict__ spacing,  // [3]
    const unsigned long long* __restrict__ ws,
    float* __restrict__ out,            // [B,4]
    int B, int n)
{
  int b = blockIdx.x * blockDim.x + threadIdx.x;
  if (b >= B) return;

  // hard part: gather metrics at the argmin index
  unsigned int idx = (unsigned int)(ws[b] & 0xFFFFFFFFull);
  out[b*4+0] = met[(size_t)idx*2+0];
  out[b*4+1] = met[(size_t)idx*2+1];

  // soft part
  float qx = q[b*3+0], qy = q[b*3+1], qz = q[b*3+2];
  float rx = (qx - origin[0]) / spacing[0];
  float ry = (qy - origin[1]) / spacing[1];
  float rz = (qz - origin[2]) / spacing[2];
  int vx = (int)rintf(rx), vy = (int)rintf(ry), vz = (int)rintf(rz);
  vx = min(max(vx, 1), n - 2);
  vy = min(max(vy, 1), n - 2);
  vz = min(max(vz, 1), n - 2);
  float t = temp_p[0] + 1e-8f;

  float ls[27];
  float maxl = -3.0e38f;
  int k = 0;
  for (int dx = -1; dx <= 1; ++dx)
    for (int dy = -1; dy <= 1; ++dy)
      for (int dz = -1; dz <= 1; ++dz) {
        size_t g = (((size_t)(vx+dx) * n) + (vy+dy)) * n + (vz+dz);
        float gx = grid[g*3+0], gy = grid[g*3+1], gz = grid[g*3+2];
        float ex = qx - gx, ey = qy - gy, ez = qz - gz;
        float d2 = ex*ex + ey*ey + ez*ez;
        float l = -d2 / t;
        ls[k++] = l;
        maxl = fmaxf(maxl, l);
      }

  float sum = 0.f, a0 = 0.f, a1 = 0.f;
  k = 0;
  for (int dx = -1; dx <= 1; ++dx)
    for (int dy = -1; dy <= 1; ++dy)
      for (int dz = -1; dz <= 1; ++dz) {
        size_t g = (((size_t)(vx+dx) * n) + (vy+dy)) * n + (vz+dz);
        float w = __expf(ls[k++] - maxl);
        sum += w;
        a0 += w * met[g*2+0];
        a1 += w * met[g*2+1];
      }
  out[b*4+2] = a0 / sum;
  out[b*4+3] = a1 / sum;
}

// ---------------------------------------------------------------------------
extern "C" void kernel_launch(void* const* d_in, const int* in_sizes, int n_in,
                              void* d_out, int out_size, void* d_ws, size_t ws_size,
                              hipStream_t stream) {
  const float* q       = (const float*)d_in[0];   // query_vectors [B,3]
  const float* temps   = (const float*)d_in[1];   // temperatures  [1]
  const float* grid    = (const float*)d_in[2];   // indices_3d    [n,n,n,3]
  const float* met     = (const float*)d_in[3];   // relevant_metrics_3d [n,n,n,2]
  const float* origin  = (const float*)d_in[4];   // [3]
  const float* spacing = (const float*)d_in[5];   // [3]
  float* out = (float*)d_out;

  const int B   = in_sizes[0] / 3;
  const int ndb = in_sizes[2] / 3;
  int n = (int)floor(cbrt((double)ndb) + 0.5);    // cubic grid (96)

  unsigned long long* ws = (unsigned long long*)d_ws;

  ll_init<<<(B + 255) / 256, 256, 0, stream>>>(ws, B);

  const int qtiles = (B + 15) / 16;
  const int gy = (qtiles + WAVES_PER_BLOCK - 1) / WAVES_PER_BLOCK;
  const int tiles = (ndb + 15) / 16;
  const int gx = (tiles + TILES_PER_WAVE - 1) / TILES_PER_WAVE;
  ll_hard_wmma<<<dim3(gx, gy), 256, 0, stream>>>(q, grid, ws, B, ndb);

  ll_finalize<<<(B + 255) / 256, 256, 0, stream>>>(
      q, temps, grid, met, origin, spacing, ws, out, B, n);
}